// AttentionModel_13005160973050
// MI455X (gfx1250) — compile-verified
//
#include <hip/hip_runtime.h>
#include <hip/hip_bf16.h>
#include <math.h>

typedef float v2f __attribute__((ext_vector_type(2)));
typedef float v8f __attribute__((ext_vector_type(8)));
typedef unsigned int v4u __attribute__((ext_vector_type(4)));
typedef unsigned int v8u __attribute__((ext_vector_type(8)));

#define B_SZ 256
#define N_SZ 2048
#define D_SZ 128

// Async copy 16B per lane: global -> LDS, tracked by ASYNCcnt
__device__ __forceinline__ void async_ld_b128(unsigned lds_off, const void* gaddr) {
  asm volatile("global_load_async_to_lds_b128 %0, %1, off"
               :: "v"(lds_off), "v"(gaddr) : "memory");
}
__device__ __forceinline__ void wait_asynccnt0() {
  asm volatile("s_wait_asynccnt 0x0" ::: "memory");
}
__device__ __forceinline__ void wait_tensorcnt0() {
#if __has_builtin(__builtin_amdgcn_s_wait_tensorcnt)
  __builtin_amdgcn_s_wait_tensorcnt(0);
#else
  asm volatile("s_wait_tensorcnt 0x0" ::: "memory");
#endif
}

// TDM: DMA one 64x128 f32 tile (row stride 128 elems) from global into LDS,
// padding LDS dest by 4 DWORDs per 128 DWORDs -> LDS row stride 132 floats.
__device__ __forceinline__ void tdm_load_tile(const void* gaddr, unsigned lds_addr) {
  const unsigned long long ga = (unsigned long long)(uintptr_t)gaddr;
  v4u g0;
  g0[0] = 1u;                                               // count=1 (valid user D#)
  g0[1] = lds_addr;                                         // lds_addr[31:0]
  g0[2] = (unsigned)ga;                                     // global_addr[31:0]
  g0[3] = (unsigned)((ga >> 32) & 0x01FFFFFFu) | (2u << 30); // addr[56:32] | type=2
  v8u g1;
  g1[0] = (2u << 16)       // data_size = 4B
        | (1u << 20)       // pad_enable
        | (6u << 22)       // pad_interval: 128 DWORDs (512B rows)
        | (3u << 25);      // pad_amount: 4 DWORDs (16B) -> stride 132 floats
  g1[1] = (128u << 16);    // tensor_dim0 = 128 (abar_addr = 0)
  g1[2] = (2048u << 16);   // tensor_dim1 = 2048 (lo16); tensor_dim0 hi = 0
  g1[3] = (128u << 16);    // tile_dim0 = 128; tensor_dim1 hi = 0
  g1[4] = 64u;             // tile_dim1 = 64; tile_dim2 = 0
  g1[5] = 128u;            // tensor_dim0_stride = 128 (lo32)
  g1[6] = 0u;              // stride hi, tensor_dim1_stride lo
  g1[7] = 0u;
  asm volatile("tensor_load_to_lds %0, %1" :: "s"(g0), "s"(g1) : "memory");
}

// ---------------------------------------------------------------------------
// Kernel 1: per-b prep.  graph mean -> fixed context -> query -> qk matrix.
// qk[k][h] = (1/sqrt(hd)) * sum_j W_node[k][16h+j] * query[16h+j]   (h<8)
// ---------------------------------------------------------------------------
__global__ __launch_bounds__(128) void prep_kernel(
    const float* __restrict__ emb, const float* __restrict__ Wfixed,
    const float* __restrict__ Wstep, const float* __restrict__ Wnode,
    const int* __restrict__ firstIdx, const int* __restrict__ lastIdx,
    float* __restrict__ qkOut)
{
  __shared__ float geL[128], feL[128], leL[128], qL[128];
  const int b = blockIdx.x, t = threadIdx.x;
  const float* eb = emb + (size_t)b * N_SZ * D_SZ;

  float s = 0.f;
  #pragma unroll 4
  for (int n = 0; n < N_SZ; ++n) s += eb[(size_t)n * D_SZ + t];
  geL[t] = s * (1.0f / (float)N_SZ);
  feL[t] = eb[(size_t)firstIdx[b] * D_SZ + t];
  leL[t] = eb[(size_t)lastIdx[b] * D_SZ + t];
  __syncthreads();

  float q = 0.f;
  for (int k = 0; k < 128; ++k) q += geL[k] * Wfixed[k * 128 + t];
  for (int k = 0; k < 128; ++k) q += feL[k] * Wstep[k * 128 + t];
  for (int k = 0; k < 128; ++k) q += leL[k] * Wstep[(128 + k) * 128 + t];
  qL[t] = q;
  __syncthreads();

  float* out = qkOut + ((size_t)b * 128 + t) * 16;
  #pragma unroll
  for (int h = 0; h < 8; ++h) {
    float a = 0.f;
    #pragma unroll
    for (int j = 0; j < 16; ++j) a += Wnode[(size_t)t * 384 + 16 * h + j] * qL[16 * h + j];
    out[h] = 0.25f * a;                       // 1/sqrt(hd) = 1/4 folded in
  }
  #pragma unroll
  for (int h = 8; h < 16; ++h) out[h] = 0.f;  // pad heads -> zero columns
}

// ---------------------------------------------------------------------------
// Kernel 2: fused glimpse attention.  E tiles staged by the Tensor Data Mover
// (one tensor_load_to_lds per 32KB tile, TENSORcnt), mask row staged via
// GLOBAL_LOAD_ASYNC_TO_LDS_B128 (ASYNCcnt), compat + weighted-embedding
// accumulation via chained V_WMMA_F32_16X16X4_F32.  One block per b.
// ---------------------------------------------------------------------------
__global__ __launch_bounds__(256) void attn_kernel(
    const float* __restrict__ emb, const float* __restrict__ Wnode,
    const float* __restrict__ Wout, const unsigned char* __restrict__ mask,
    const float* __restrict__ qkIn, float* __restrict__ gvecOut)
{
  __shared__ float emsL[64 * 132];   // E supertile [64 nodes][128 dims], padded
  __shared__ float cpL[8 * 66];      // compat / p, [8 heads][64 nodes], padded
  __shared__ float aL[8 * 128];      // normalized weighted embeddings
  __shared__ float scaleL[8];
  __shared__ float sArr[8];
  __shared__ float gL[128], glL[128];
  __shared__ __align__(16) unsigned char mkAll[N_SZ];  // full mask row for b

  const int b = blockIdx.x;
  const int tid = threadIdx.x;
  const int w = tid >> 5;       // wave id 0..7
  const int lane = tid & 31;
  const int lm = lane & 15;
  const int lh = lane >> 4;

  const float* eb = emb + (size_t)b * N_SZ * D_SZ;
  const unsigned char* mb = mask + (size_t)b * N_SZ;
  const unsigned emsBase = (unsigned)(uintptr_t)(const void*)&emsL[0];

  // whole mask row -> LDS via async-LDS loads (16B per thread, threads 0..127)
  if (tid < 128) {
    async_ld_b128((unsigned)(uintptr_t)(void*)&mkAll[tid * 16], mb + tid * 16);
  }

  // Preload the qk B-matrix fragments (only waves 0..3 run compat).
  v2f b1f[32];
  if (w < 4) {
    const float* qk = qkIn + (size_t)b * 128 * 16;
    #pragma unroll
    for (int kk = 0; kk < 32; ++kk) {
      const int r = 4 * kk + 2 * lh;
      b1f[kk][0] = qk[r * 16 + lm];
      b1f[kk][1] = qk[(r + 1) * 16 + lm];
    }
  }
  wait_asynccnt0();

  v8f acc = {0.f, 0.f, 0.f, 0.f, 0.f, 0.f, 0.f, 0.f};
  float m = -1e30f, ssum = 0.f;

  for (int t = 0; t < 32; ++t) {            // 32 supertiles of 64 nodes
    __syncthreads();                        // all readers of emsL/cpL done
    // warm L2 for the next tile while this tile's DMA runs
    if (t + 1 < 32) {
      __builtin_prefetch(eb + (size_t)(t + 1) * 64 * D_SZ + tid * 32, 0, 0);
    }
    if (w == 0) {                           // one TDM descriptor per tile
      tdm_load_tile(eb + (size_t)t * 64 * D_SZ, emsBase);
      wait_tensorcnt0();
    }
    __syncthreads();

    // --- compat GEMM: waves 0..3, subtile w (16 nodes) x 16 cols (8 heads) ---
    if (w < 4) {
      v8f c = {0.f, 0.f, 0.f, 0.f, 0.f, 0.f, 0.f, 0.f};
      #pragma unroll
      for (int kk = 0; kk < 32; ++kk) {
        const int col = 4 * kk + 2 * lh;
        v2f a;
        a[0] = emsL[(16 * w + lm) * 132 + col];
        a[1] = emsL[(16 * w + lm) * 132 + col + 1];
        c = __builtin_amdgcn_wmma_f32_16x16x4_f32(false, a, false, b1f[kk],
                                                  (short)0, c, false, false);
      }
      if (lm < 8) {                     // cols 0..7 = real heads
        #pragma unroll
        for (int v = 0; v < 8; ++v) {
          const int node = 16 * w + v + 8 * lh;
          cpL[lm * 66 + node] = mkAll[t * 64 + node] ? -1e30f : c[v];
        }
      }
    }
    __syncthreads();

    // --- online softmax: wave w owns head w (64 nodes) ---
    {
      float c0 = cpL[w * 66 + lane];
      float c1 = cpL[w * 66 + lane + 32];
      float tmax = fmaxf(c0, c1);
      #pragma unroll
      for (int d = 16; d >= 1; d >>= 1) tmax = fmaxf(tmax, __shfl_xor(tmax, d, 32));
      const float mn = fmaxf(m, tmax);
      const float sc = expf(m - mn);
      const float p0 = expf(c0 - mn);
      const float p1 = expf(c1 - mn);
      cpL[w * 66 + lane] = p0;
      cpL[w * 66 + lane + 32] = p1;
      float ps = p0 + p1;
      #pragma unroll
      for (int d = 16; d >= 1; d >>= 1) ps += __shfl_xor(ps, d, 32);
      ssum = ssum * sc + ps;
      m = mn;
      if (lane == 0) scaleL[w] = sc;
    }
    __syncthreads();

    // --- accumulation GEMM: acc[16 heads x 16 dims] += P x E, K = 64 nodes ---
    {
      #pragma unroll
      for (int v = 0; v < 8; ++v) acc[v] *= scaleL[(v + 8 * lh) & 7];
      #pragma unroll
      for (int kk = 0; kk < 16; ++kk) {
        const int node = 4 * kk + 2 * lh;
        v2f a;
        if (lm < 8) { a[0] = cpL[lm * 66 + node]; a[1] = cpL[lm * 66 + node + 1]; }
        else        { a[0] = 0.f;                 a[1] = 0.f; }
        v2f bb;
        bb[0] = emsL[node * 132 + 16 * w + lm];
        bb[1] = emsL[(node + 1) * 132 + 16 * w + lm];
        acc = __builtin_amdgcn_wmma_f32_16x16x4_f32(false, a, false, bb,
                                                    (short)0, acc, false, false);
      }
    }
  }

  if (lane == 0) sArr[w] = ssum;
  __syncthreads();
  if (lane < 16) {
    #pragma unroll
    for (int v = 0; v < 8; ++v) aL[v * 128 + 16 * w + lane] = acc[v] / sArr[v];
  }
  __syncthreads();

  // heads -> glimpse -> gvec (all tiny, scalar)
  if (tid < 128) {
    const int h = tid >> 4;
    float g = 0.f;
    for (int k = 0; k < 128; ++k) g += aL[h * 128 + k] * Wnode[(size_t)k * 384 + 128 + tid];
    gL[tid] = g;
  }
  __syncthreads();
  if (tid < 128) {
    float gl = 0.f;
    for (int j = 0; j < 128; ++j) gl += gL[j] * Wout[j * 128 + tid];
    glL[tid] = gl;
  }
  __syncthreads();
  if (tid < 128) {
    float gv = 0.f;
    for (int d = 0; d < 128; ++d) gv += Wnode[(size_t)tid * 384 + 256 + d] * glL[d];
    gvecOut[(size_t)b * 128 + tid] = gv * 0.08838834764831845f;   // 1/sqrt(128)
  }
}

// ---------------------------------------------------------------------------
// Kernel 3: logits GEMV + tanh clip + mask + log_softmax.  One block per b.
// ---------------------------------------------------------------------------
__global__ __launch_bounds__(256) void logits_kernel(
    const float* __restrict__ emb, const unsigned char* __restrict__ mask,
    const float* __restrict__ gvec, float* __restrict__ out)
{
  __shared__ float gvL[128];
  __shared__ float red[8];
  const int b = blockIdx.x, tid = threadIdx.x, w = tid >> 5, lane = tid & 31;
  if (tid < 128) gvL[tid] = gvec[(size_t)b * 128 + tid];
  __syncthreads();

  const float* eb = emb + (size_t)b * N_SZ * D_SZ;
  const unsigned char* mb = mask + (size_t)b * N_SZ;
  float lv[8];

  for (int i = 0; i < 256; ++i) {
    const int r = w * 256 + i;
    const float4 e = ((const float4*)(eb + (size_t)r * D_SZ))[lane];
    float p = e.x * gvL[4 * lane] + e.y * gvL[4 * lane + 1] +
              e.z * gvL[4 * lane + 2] + e.w * gvL[4 * lane + 3];
    #pragma unroll
    for (int d = 16; d >= 1; d >>= 1) p += __shfl_xor(p, d, 32);
    float lg = tanhf(p) * 10.0f;
    if (mb[r]) lg = -INFINITY;
    if ((i & 31) == lane) lv[i >> 5] = lg;
  }

  float mx = lv[0];
  #pragma unroll
  for (int j = 1; j < 8; ++j) mx = fmaxf(mx, lv[j]);
  #pragma unroll
  for (int d = 16; d >= 1; d >>= 1) mx = fmaxf(mx, __shfl_xor(mx, d, 32));
  if (lane == 0) red[w] = mx;
  __syncthreads();
  float bm = red[0];
  #pragma unroll
  for (int j = 1; j < 8; ++j) bm = fmaxf(bm, red[j]);
  __syncthreads();

  float es = 0.f;
  #pragma unroll
  for (int j = 0; j < 8; ++j) es += expf(lv[j] - bm);
  #pragma unroll
  for (int d = 16; d >= 1; d >>= 1) es += __shfl_xor(es, d, 32);
  if (lane == 0) red[w] = es;
  __syncthreads();
  float bs = 0.f;
  #pragma unroll
  for (int j = 0; j < 8; ++j) bs += red[j];
  const float lz = bm + logf(bs);

  float* ob = out + (size_t)b * N_SZ;
  #pragma unroll
  for (int j = 0; j < 8; ++j) ob[w * 256 + j * 32 + lane] = lv[j] - lz;
}

// ---------------------------------------------------------------------------
extern "C" void kernel_launch(void* const* d_in, const int* in_sizes, int n_in,
                              void* d_out, int out_size, void* d_ws, size_t ws_size,
                              hipStream_t stream) {
  const float* emb    = (const float*)d_in[0];   // [B,N,D]
  const float* Wnode  = (const float*)d_in[1];   // [D,3D]
  const float* Wfixed = (const float*)d_in[2];   // [D,D]
  const float* Wstep  = (const float*)d_in[3];   // [2D,D]
  const float* Wout   = (const float*)d_in[4];   // [D,D]
  const int*   fi     = (const int*)d_in[5];     // [B]
  const int*   li     = (const int*)d_in[6];     // [B]
  const unsigned char* mask = (const unsigned char*)d_in[7];  // [B,N] bool
  float* out = (float*)d_out;

  float* gvec = (float*)d_ws;                    // B*128 floats
  float* qk   = gvec + (size_t)B_SZ * 128;       // B*128*16 floats

  prep_kernel<<<B_SZ, 128, 0, stream>>>(emb, Wfixed, Wstep, Wnode, fi, li, qk);
  attn_kernel<<<B_SZ, 256, 0, stream>>>(emb, Wnode, Wout, mask, qk, gvec);
  logits_kernel<<<B_SZ, 256, 0, stream>>>(emb, mask, gvec, out);
}